// AbstractionLayer_84267258347623
// MI455X (gfx1250) — compile-verified
//
#include <hip/hip_runtime.h>

typedef __attribute__((ext_vector_type(2))) float v2f;
typedef __attribute__((ext_vector_type(8))) float v8f;

#define BATCH   524288
#define NI      12
#define NR      6
#define NJ      2
#define NL      2
#define NV      4
#define NRJ     (NR*NJ)            // 12
#define TILE_B  16
#define NTILES  (BATCH / TILE_B)   // 32768
#define FPB     (NI * NL)          // 24 floats per batch element
#define LOG2E   1.4426950408889634f

__global__ __launch_bounds__(256, 6) void
abstraction_layer_wmma_kernel(const float* __restrict__ cf,
                              const float* __restrict__ templates,
                              const float* __restrict__ gammas,
                              const float* __restrict__ body_W,
                              const float* __restrict__ body_b,
                              const float* __restrict__ head_W,
                              const float* __restrict__ head_b,
                              float* __restrict__ out)
{
    // One private 1536B LDS slab per wave (8 waves / 256-thread block = 12 KB).
    __shared__ float lds[8][TILE_B * FPB];
    const int lane = threadIdx.x & 31;
    const int wv   = threadIdx.x >> 5;
    float* myLds = lds[wv];

    const int  n     = lane & 15;      // D/B/C column: rj index this lane owns
    const bool hiK   = lane >= 16;     // lanes 16..31 hold K=2,3 of A/B, rows m=8..15 of D
    const int  mOff  = hiK ? 8 : 0;
    const bool valid = (n < NRJ);
    const int  nc    = valid ? n : (NRJ - 1);
    const int  r     = nc >> 1;

    // ---------------- per-lane constant folding ----------------
    // w = 1 - clamp(gamma,0,1); logits are linear in (f, f^2), so fold log2(e)
    // into B and C: D is then base-2 logits and exp() is a bare v_exp_f32.
    // B[k,n] = log2e * {2*w*t (k=0,1), -w (k=2,3)};  C = -log2e * c0
    float t0 = templates[nc*NL+0], t1 = templates[nc*NL+1];
    float g0 = gammas[nc*NL+0],    g1 = gammas[nc*NL+1];
    g0 = fminf(fmaxf(g0, 0.f), 1.f);
    g1 = fminf(fmaxf(g1, 0.f), 1.f);
    const float w0 = 1.f - g0, w1 = 1.f - g1;
    const float c0 = w0*t0*t0 + w1*t1*t1;
    v2f Bm;
    Bm.x = valid ? LOG2E * (hiK ? -w0 : 2.f*w0*t0) : 0.f;
    Bm.y = valid ? LOG2E * (hiK ? -w1 : 2.f*w1*t1) : 0.f;
    const float cval = valid ? -LOG2E * c0 : 0.f;

    // Fold body (per r,j) and head (per r) linears: M2[l][l'] = sum_v headW[r,l,v]*bodyW[r,j,v,l']
    // bias[l] = head_b[r,l] + sum_v headW[r,l,v]*(body_b[r,0,v]+body_b[r,1,v])
    float m00 = 0.f, m01 = 0.f, m10 = 0.f, m11 = 0.f;
    float bias0 = head_b[r*NL+0], bias1 = head_b[r*NL+1];
#pragma unroll
    for (int v = 0; v < NV; ++v) {
        const float hw0 = head_W[(r*NL+0)*NV + v];
        const float hw1 = head_W[(r*NL+1)*NV + v];
        const float bw0 = body_W[(nc*NV+v)*NL + 0];
        const float bw1 = body_W[(nc*NV+v)*NL + 1];
        m00 += hw0*bw0; m01 += hw0*bw1;
        m10 += hw1*bw0; m11 += hw1*bw1;
        const float sb = body_b[(r*NJ+0)*NV + v] + body_b[(r*NJ+1)*NV + v];
        bias0 += hw0 * sb;
        bias1 += hw1 * sb;
    }

    const int waveId = blockIdx.x * (blockDim.x >> 5) + wv;
    const int nWaves = gridDim.x * (blockDim.x >> 5);

    for (int t = waveId; t < NTILES; t += nWaves) {
        const int base = t * TILE_B;

        // ---- stage tile features (16 x 12 x 2 f32 = 1536 B) into our LDS slab, coalesced
        {
            const float4* src = reinterpret_cast<const float4*>(cf + (size_t)base * FPB);
            float4* dst = reinterpret_cast<float4*>(myLds);
#pragma unroll
            for (int k = 0; k < (TILE_B*FPB/4)/32; ++k)
                dst[lane + 32*k] = src[lane + 32*k];
        }
        // prefetch the next tile this wave will process
        if (t + nWaves < NTILES)
            __builtin_prefetch(cf + (size_t)(base + nWaves*TILE_B) * FPB, 0, 1);
        // make our cross-lane LDS writes visible before row-major reads
        asm volatile("s_wait_dscnt 0x0" ::: "memory");

        float den[8], num0[8], num1[8];
#pragma unroll
        for (int v = 0; v < 8; ++v) { den[v] = 0.f; num0[v] = 0.f; num1[v] = 0.f; }

        // Partial unroll: 4 WMMA tiles in flight is enough ILP to overlap the
        // matrix pipe with v_exp_f32, while keeping VGPRs < 256 (no MSB banking).
#pragma unroll 4
        for (int i = 0; i < NI; ++i) {
            // A row m = lane&15 : K = {f0, f1, f0^2, f1^2}
            const float2 fa = *reinterpret_cast<const float2*>(&myLds[n*FPB + i*NL]);
            v2f A;
            A.x = hiK ? fa.x*fa.x : fa.x;
            A.y = hiK ? fa.y*fa.y : fa.y;
            v8f C;
#pragma unroll
            for (int e = 0; e < 8; ++e) C[e] = cval;
            // D[m=b_local][n=rj] = -log2e * match_score  (base-2 softmax logits)
            v8f D = __builtin_amdgcn_wmma_f32_16x16x4_f32(
                        false, A, false, Bm, (short)0, C, false, false);
            // streaming (unnormalized) softmax accumulation; logits bounded, no max needed
#pragma unroll
            for (int v = 0; v < 8; ++v) {
                const float e = __builtin_amdgcn_exp2f(D[v]);   // bare v_exp_f32
                const float2 f = *reinterpret_cast<const float2*>(
                                     &myLds[(v+mOff)*FPB + i*NL]);  // LDS broadcast read
                den[v]  += e;
                num0[v] += e * f.x;
                num1[v] += e * f.y;
            }
        }

        // ---- selected -> folded 2x2 linear; combine j=0/j=1 across adjacent lanes; store
#pragma unroll
        for (int v = 0; v < 8; ++v) {
            const float inv = __builtin_amdgcn_rcpf(den[v]);   // v_rcp_f32, ~1ulp
            const float s0 = num0[v] * inv;
            const float s1 = num1[v] * inv;
            float p0 = m00*s0 + m01*s1;     // this lane's (r,j) partial for l=0
            float p1 = m10*s0 + m11*s1;     // ... for l=1
            const float q0 = __shfl_xor(p0, 1, 32);  // partner template j^1
            const float q1 = __shfl_xor(p1, 1, 32);
            if (valid && !(n & 1)) {
                const int b = base + v + mOff;
                float2 o;
                o.x = p0 + q0 + bias0;
                o.y = p1 + q1 + bias1;
                reinterpret_cast<float2*>(out)[(size_t)b*NR + r] = o;
            }
        }
    }
}

extern "C" void kernel_launch(void* const* d_in, const int* in_sizes, int n_in,
                              void* d_out, int out_size, void* d_ws, size_t ws_size,
                              hipStream_t stream) {
    const float* cf        = (const float*)d_in[0];
    const float* templates = (const float*)d_in[1];
    const float* gammas    = (const float*)d_in[2];
    const float* body_W    = (const float*)d_in[3];
    const float* body_b    = (const float*)d_in[4];
    const float* head_W    = (const float*)d_in[5];
    const float* head_b    = (const float*)d_in[6];
    float* out = (float*)d_out;

    dim3 grid(1024), block(256);   // 8192 waves, 4 tiles of 16 batch elements each
    abstraction_layer_wmma_kernel<<<grid, block, 0, stream>>>(
        cf, templates, gammas, body_W, body_b, head_W, head_b, out);
}